// UserItemWithRepostTimeEmbedding_70755291234321
// MI455X (gfx1250) — compile-verified
//
#include <hip/hip_runtime.h>
#include <hip/hip_bf16.h>
#include <cstddef>

#define NTOKEN 150000
#define NINP   64
#define NHID   128
#define NEDGE  2400000
#define BATCH  64
#define MAXLEN 200

typedef __attribute__((ext_vector_type(2))) float v2f;
typedef __attribute__((ext_vector_type(8))) float v8f;

// ----------------------------------------------------------------------------
// Degree computation: deg = 1 (self loop) + in-degree(dst); then dinv = rsqrt
// ----------------------------------------------------------------------------
__global__ void k_deg_init(float* __restrict__ deg) {
    int i = blockIdx.x * blockDim.x + threadIdx.x;
    if (i < NTOKEN) deg[i] = 1.0f;  // self loop
}

__global__ void k_deg_count(const int* __restrict__ ei, float* __restrict__ deg) {
    int e = blockIdx.x * blockDim.x + threadIdx.x;
    if (e < NEDGE) atomicAdd(&deg[ei[NEDGE + e]], 1.0f);  // dst row of edge_index
}

__global__ void k_deg_rsqrt(float* __restrict__ deg) {
    int i = blockIdx.x * blockDim.x + threadIdx.x;
    if (i < NTOKEN) deg[i] = rsqrtf(deg[i]);
}

// ----------------------------------------------------------------------------
// GEMM  C[M=NTOKEN, N] = A[M, K] * B[K, N]  via V_WMMA_F32_16X16X4_F32.
// One wave per 16-row tile, covering all NTILES*16 columns (acc stays in VGPRs).
// A frag (16x4 fp32): lane 0-15 -> M=lane, K=k0+{0,1}; lane 16-31 -> K=k0+{2,3}.
// B frag (4x16): VGPR0 = rows K=k0 (lanes 0-15) / K=k0+2 (lanes 16-31), VGPR1 = +1.
// C/D (16x16 f32): VGPR v -> row v (lanes 0-15) / row v+8 (lanes 16-31).
// ----------------------------------------------------------------------------
template<int K, int NTILES>
__global__ void k_gemm_wmma(const float* __restrict__ A, const float* __restrict__ B,
                            float* __restrict__ C) {
    const int N     = NTILES * 16;
    const int lane  = threadIdx.x & 31;
    const int mtile = blockIdx.x * (blockDim.x >> 5) + (threadIdx.x >> 5);
    const int m0    = mtile * 16;
    if (m0 >= NTOKEN) return;   // whole-wave exit: EXEC all-1s for surviving waves

    const v8f zero = {0.f, 0.f, 0.f, 0.f, 0.f, 0.f, 0.f, 0.f};
    v8f acc[NTILES];
#pragma unroll
    for (int t = 0; t < NTILES; ++t) acc[t] = zero;

    const int row = m0 + (lane & 15);
    const int kh  = (lane >> 4) << 1;          // 0 for lanes 0-15, 2 for 16-31
    const int col = lane & 15;
    const float* __restrict__ Arow = A + (size_t)row * K;

#pragma unroll 4
    for (int k0 = 0; k0 < K; k0 += 4) {
        v2f a;
        a.x = Arow[k0 + kh];
        a.y = Arow[k0 + kh + 1];
        const float* __restrict__ B0 = B + (size_t)(k0 + kh) * N;
#pragma unroll
        for (int t = 0; t < NTILES; ++t) {
            v2f b;
            b.x = B0[t * 16 + col];
            b.y = B0[N + t * 16 + col];
            acc[t] = __builtin_amdgcn_wmma_f32_16x16x4_f32(
                false, a, false, b, (short)0, acc[t], false, false);
        }
    }

    const int rbase = m0 + ((lane >> 4) << 3);  // lanes 16-31 hold rows +8
#pragma unroll
    for (int t = 0; t < NTILES; ++t) {
#pragma unroll
        for (int v = 0; v < 8; ++v) {
            C[(size_t)(rbase + v) * N + t * 16 + col] = acc[t][v];
        }
    }
}

// ----------------------------------------------------------------------------
// out[i,:] = xw[i,:] * dinv[i]^2 + bias[:]   (self-loop term + bias; also
// fully initializes the output so the edge-scatter atomics start from it)
// ----------------------------------------------------------------------------
template<int F>
__global__ void k_self_bias(const float* __restrict__ x, const float* __restrict__ dinv,
                            const float* __restrict__ bias, float* __restrict__ out) {
    size_t i = (size_t)blockIdx.x * blockDim.x + threadIdx.x;
    if (i >= (size_t)NTOKEN * F) return;
    const int node = (int)(i / F);
    const int f    = (int)(i % F);
    const float d  = dinv[node];
    out[i] = x[i] * d * d + bias[f];
}

// ----------------------------------------------------------------------------
// Edge scatter: out[dst,:] += xw[src,:] * dinv[src]*dinv[dst]
// One wave32 per edge; lane handles F/32 features (F = 128 or 64).
// ----------------------------------------------------------------------------
template<int F>
__global__ void k_edge_scatter(const int* __restrict__ ei, const float* __restrict__ x,
                               const float* __restrict__ dinv, float* __restrict__ out) {
    const int lane = threadIdx.x & 31;
    const int e    = blockIdx.x * (blockDim.x >> 5) + (threadIdx.x >> 5);
    if (e >= NEDGE) return;
    const int src = ei[e];
    const int dst = ei[NEDGE + e];
    const float norm = dinv[src] * dinv[dst];
    const float* __restrict__ xs = x + (size_t)src * F;
    float* __restrict__ od = out + (size_t)dst * F;
#pragma unroll
    for (int f = lane; f < F; f += 32) {
        atomicAdd(&od[f], xs[f] * norm);
    }
}

// ----------------------------------------------------------------------------
// Output gather: out[b,l,:] = nodes[input[b,l], :]
// ----------------------------------------------------------------------------
__global__ void k_gather(const int* __restrict__ idx, const float* __restrict__ nodes,
                         float* __restrict__ out) {
    const int total = BATCH * MAXLEN * NINP;
    int i = blockIdx.x * blockDim.x + threadIdx.x;
    if (i >= total) return;
    const int tok = i >> 6;   // / NINP
    const int f   = i & 63;   // % NINP
    out[i] = nodes[(size_t)idx[tok] * NINP + f];
}

// ----------------------------------------------------------------------------
extern "C" void kernel_launch(void* const* d_in, const int* in_sizes, int n_in,
                              void* d_out, int out_size, void* d_ws, size_t ws_size,
                              hipStream_t stream) {
    const float* emb = (const float*)d_in[0];   // [NTOKEN, 64]
    const float* W1  = (const float*)d_in[1];   // [64, 128]
    const float* b1  = (const float*)d_in[2];   // [128]
    const float* W2  = (const float*)d_in[3];   // [128, 64]
    const float* b2  = (const float*)d_in[4];   // [64]
    const int*   inp = (const int*)d_in[5];     // [64, 200]
    // d_in[6] = input_timestamp (unused by reference output)
    const int*   ei  = (const int*)d_in[7];     // [2, NEDGE]
    float* out = (float*)d_out;                 // [64, 200, 64] fp32

    // Workspace layout (all 16B-aligned offsets):
    char* ws = (char*)d_ws;
    float* dinv  = (float*)(ws);                                          // NTOKEN
    float* xw    = (float*)(ws + (size_t)NTOKEN * 4);                     // NTOKEN*NHID (reused)
    float* h     = (float*)(ws + (size_t)NTOKEN * 4
                               + (size_t)NTOKEN * NHID * 4);              // NTOKEN*NHID
    float* nodes = (float*)(ws + (size_t)NTOKEN * 4
                               + (size_t)NTOKEN * NHID * 8);              // NTOKEN*NINP

    // --- degrees -> dinv ---
    k_deg_init <<<(NTOKEN + 255) / 256, 256, 0, stream>>>(dinv);
    k_deg_count<<<(NEDGE  + 255) / 256, 256, 0, stream>>>(ei, dinv);
    k_deg_rsqrt<<<(NTOKEN + 255) / 256, 256, 0, stream>>>(dinv);

    const int WAVES  = 4;                 // 4 waves / 128-thread block
    const int mtiles = NTOKEN / 16;       // 9375

    // --- layer 1: h = P * (emb @ W1) + b1 ---
    k_gemm_wmma<64, 8><<<(mtiles + WAVES - 1) / WAVES, WAVES * 32, 0, stream>>>(emb, W1, xw);
    k_self_bias<NHID><<<(int)(((size_t)NTOKEN * NHID + 255) / 256), 256, 0, stream>>>(xw, dinv, b1, h);
    k_edge_scatter<NHID><<<(NEDGE + 7) / 8, 256, 0, stream>>>(ei, xw, dinv, h);

    // --- layer 2: nodes = P * (h @ W2) + b2   (xw buffer reused for h@W2) ---
    k_gemm_wmma<128, 4><<<(mtiles + WAVES - 1) / WAVES, WAVES * 32, 0, stream>>>(h, W2, xw);
    k_self_bias<NINP><<<(int)(((size_t)NTOKEN * NINP + 255) / 256), 256, 0, stream>>>(xw, dinv, b2, nodes);
    k_edge_scatter<NINP><<<(NEDGE + 7) / 8, 256, 0, stream>>>(ei, xw, dinv, nodes);

    // --- gather embeddings for the batch ---
    k_gather<<<(BATCH * MAXLEN * NINP + 255) / 256, 256, 0, stream>>>(inp, nodes, out);
}